// FilterGeneratingNetwork_18648747999355
// MI455X (gfx1250) — compile-verified
//
#include <hip/hip_runtime.h>
#include <hip/hip_bf16.h>

typedef __attribute__((ext_vector_type(2))) float v2f;
typedef __attribute__((ext_vector_type(8))) float v8f;

#define HW   (256 * 256)
#define CIN  32
#define COUT 32
#define NB   8

// ---------------------------------------------------------------------------
// Kernel 1: 1x1 conv (channel mixing) via V_WMMA_F32_16X16X4_F32.
// feat[b,o,p] = sum_i w[o,i] * x[b,i,p] + bias[o]
// Each wave: 32 output channels x 16 contiguous pixels.
//   A (16x4 f32): lane%16 = M (=o within tile); VGPR0/1 hold K / K+1,
//                 upper half-wave holds K+2 / K+3.
//   B (4x16 f32): lanes 0-15 = N (pixel), VGPR0/1 hold rows K / K+1,
//                 upper half-wave rows K+2 / K+3.
//   C/D (16x16):  VGPR r -> M = r (lanes 0-15) / r+8 (lanes 16-31), N = lane%16.
// ---------------------------------------------------------------------------
__global__ void __launch_bounds__(256)
k_channel_mix_wmma(const float* __restrict__ x, const float* __restrict__ w,
                   const float* __restrict__ bias, float* __restrict__ feat)
{
    const int lane = threadIdx.x & 31;
    const int wave = threadIdx.x >> 5;                  // 0..7
    const int tile = (int)(blockIdx.x * 8u) + wave;     // 0..32767
    const int b    = tile >> 12;                        // tile / 4096
    const int p0   = (tile & 4095) << 4;                // base pixel
    const int half = lane >> 4;                         // 0 | 1
    const int l16  = lane & 15;

    // Preload the full 32x32 weight matrix as WMMA A-operands (2 o-tiles x 8 K-steps).
    v2f A[2][8];
#pragma unroll
    for (int t = 0; t < 2; ++t) {
        const int o = 16 * t + l16;
#pragma unroll
        for (int s = 0; s < 8; ++s) {
            const int i0 = 4 * s + 2 * half;
            A[t][s].x = w[o * CIN + i0];
            A[t][s].y = w[o * CIN + i0 + 1];
        }
    }

    // Bias-initialized accumulators (C/D layout: row r -> o = 16t + r + 8*half).
    v8f acc[2];
#pragma unroll
    for (int t = 0; t < 2; ++t)
#pragma unroll
        for (int r = 0; r < 8; ++r)
            acc[t][r] = bias[16 * t + r + 8 * half];

    const float* xb = x + (long)b * CIN * HW;
#pragma unroll
    for (int s = 0; s < 8; ++s) {
        const int i0 = 4 * s + 2 * half;
        v2f Bv;
        Bv.x = xb[(long)i0       * HW + p0 + l16];
        Bv.y = xb[(long)(i0 + 1) * HW + p0 + l16];
#pragma unroll
        for (int t = 0; t < 2; ++t)
            acc[t] = __builtin_amdgcn_wmma_f32_16x16x4_f32(
                false, A[t][s], false, Bv, (short)0, acc[t], false, false);
    }

    float* fb = feat + (long)b * COUT * HW;
#pragma unroll
    for (int t = 0; t < 2; ++t)
#pragma unroll
        for (int r = 0; r < 8; ++r)
            fb[(long)(16 * t + r + 8 * half) * HW + p0 + l16] = acc[t][r];
}

// ---------------------------------------------------------------------------
// Kernel 2: adaptive avg pool 256->5. Every bin is exactly 52 wide:
// starts {0,51,102,153,204}. filt[bc,kh,kw] = mean over 52x52 window.
// One block per (bc, kh, kw) bin; 6400 blocks.
// ---------------------------------------------------------------------------
__global__ void __launch_bounds__(256)
k_pool(const float* __restrict__ feat, float* __restrict__ filt)
{
    int id = blockIdx.x;
    const int kw = id % 5; id /= 5;
    const int kh = id % 5; id /= 5;
    const int bc = id;                     // 0..255

    const int starts[5] = {0, 51, 102, 153, 204};
    const int sh = starts[kh], sw = starts[kw];

    const float* f = feat + (long)bc * HW;
    float sum = 0.f;
    for (int idx = threadIdx.x; idx < 52 * 52; idx += 256) {
        const int r = idx / 52, c = idx % 52;
        sum += f[(sh + r) * 256 + (sw + c)];
    }

    __shared__ float red[256];
    red[threadIdx.x] = sum;
    __syncthreads();
#pragma unroll
    for (int off = 128; off > 0; off >>= 1) {
        if ((int)threadIdx.x < off) red[threadIdx.x] += red[threadIdx.x + off];
        __syncthreads();
    }
    if (threadIdx.x == 0)
        filt[(long)bc * 25 + kh * 5 + kw] = red[0] * (1.0f / (52.0f * 52.0f));
}

// ---------------------------------------------------------------------------
// Kernel 3: depthwise 5x5 correlation, pad 2 (zero). LDS-tiled 32x32 output,
// 36x36 haloed input tile (stride 37 to avoid bank conflicts).
// out[h,w] = sum_{kh,kw} filt[kh,kw] * feat[h+kh-2, w+kw-2]
// ---------------------------------------------------------------------------
__global__ void __launch_bounds__(256)
k_dwconv(const float* __restrict__ feat, const float* __restrict__ filt,
         float* __restrict__ out)
{
    const int id   = blockIdx.x;
    const int tile = id & 63;              // 8x8 tiles of 32x32
    const int bc   = id >> 6;
    const int th   = (tile >> 3) << 5;
    const int tw   = (tile & 7) << 5;

    const float* f = feat + (long)bc * HW;

    __shared__ float sF[25];
    __shared__ float sT[36][37];

    if (threadIdx.x < 25) sF[threadIdx.x] = filt[(long)bc * 25 + threadIdx.x];

    for (int i = threadIdx.x; i < 36 * 36; i += 256) {
        const int r = i / 36, c = i % 36;
        const int h = th + r - 2, w = tw + c - 2;
        float v = 0.f;
        if (h >= 0 && h < 256 && w >= 0 && w < 256) v = f[h * 256 + w];
        sT[r][c] = v;
    }
    __syncthreads();

    const int c  = threadIdx.x & 31;
    const int r0 = threadIdx.x >> 5;       // 0..7
    float* ob = out + (long)bc * HW;
#pragma unroll
    for (int rr = 0; rr < 4; ++rr) {
        const int r = r0 + rr * 8;
        float acc = 0.f;
#pragma unroll
        for (int kh = 0; kh < 5; ++kh)
#pragma unroll
            for (int kw = 0; kw < 5; ++kw)
                acc = fmaf(sF[kh * 5 + kw], sT[r + kh][c + kw], acc);
        ob[(th + r) * 256 + (tw + c)] = acc;
    }
}

// ---------------------------------------------------------------------------
extern "C" void kernel_launch(void* const* d_in, const int* in_sizes, int n_in,
                              void* d_out, int out_size, void* d_ws, size_t ws_size,
                              hipStream_t stream) {
    const float* x      = (const float*)d_in[0];   // [8,32,256,256]
    const float* conv_w = (const float*)d_in[1];   // [32,32]
    const float* conv_b = (const float*)d_in[2];   // [32]
    float*       outp   = (float*)d_out;           // [8,32,256,256]

    float* feat = (float*)d_ws;                    // 8*32*65536 floats = 64 MiB
    float* filt = feat + (long)NB * COUT * HW;     // 8*32*25 floats

    // K1: 32768 wave-tiles (16 px each), 8 waves per block -> 4096 blocks.
    k_channel_mix_wmma<<<4096, 256, 0, stream>>>(x, conv_w, conv_b, feat);
    // K2: one block per (bc, kh, kw) bin.
    k_pool<<<NB * COUT * 25, 256, 0, stream>>>(feat, filt);
    // K3: one block per (bc, 32x32 tile).
    k_dwconv<<<NB * COUT * 64, 256, 0, stream>>>(feat, filt, outp);
}